// MultiHeadGAT_43181601194640
// MI455X (gfx1250) — compile-verified
//
#include <hip/hip_runtime.h>
#include <hip/hip_bf16.h>

#define N_HEADS 8
#define HID     64
#define TOPK    16
#define NNODE   512
#define BS      32
#define DIM     128
#define E_COLS  (N_HEADS * HID)   // 512
#define LEAKY   0.2f

typedef __attribute__((ext_vector_type(2))) float v2f;
typedef __attribute__((ext_vector_type(8))) float v8f;

// ---------------------------------------------------------------------------
// Kernel 1: h_prime = h @ W via V_WMMA_F32_16X16X4_F32 (full f32 precision).
// Grid: (BS, NNODE/16). Block: 256 threads = 8 waves; wave w handles head w.
// Each wave computes a 16(nodes) x 64(hid) tile, K = 128 in steps of 4.
//
// A-frag (16x4 f32, ISA 7.12.2): lane l<16 -> row M=l, VGPR0=K0,VGPR1=K1;
//                                lane l>=16 -> row M=l-16, VGPR0=K2,VGPR1=K3.
// B-frag (4x16 f32, mirrored):   lane l<16 -> col N=l, K{0,1}; l>=16 -> K{2,3}.
// C/D (16x16 f32, 8 VGPRs): vgpr r, lane l -> M = r + 8*(l/16), N = l%16.
// ---------------------------------------------------------------------------
__global__ __launch_bounds__(256) void gat_gemm_hprime(
    const float* __restrict__ h, const float* __restrict__ W,
    float* __restrict__ hp)
{
    const int b    = blockIdx.x;
    const int rt   = blockIdx.y;
    const int hd   = threadIdx.x >> 5;   // wave id == head
    const int lane = threadIdx.x & 31;
    const int half = lane >> 4;
    const int lm   = lane & 15;

    const float* hrow   = h + ((size_t)b * NNODE + (size_t)rt * 16 + lm) * DIM;
    const int    colb   = hd * HID + lm;

    v8f acc[4] = {};
#pragma unroll 4
    for (int k4 = 0; k4 < DIM; k4 += 4) {
        const int kk = k4 + 2 * half;
        v2f a;
        a.x = hrow[kk];
        a.y = hrow[kk + 1];
#pragma unroll
        for (int nt = 0; nt < 4; ++nt) {
            const int col = colb + nt * 16;
            v2f bf;
            bf.x = W[(size_t)kk       * E_COLS + col];
            bf.y = W[(size_t)(kk + 1) * E_COLS + col];
            acc[nt] = __builtin_amdgcn_wmma_f32_16x16x4_f32(
                false, a, false, bf, (short)0, acc[nt], false, false);
        }
    }

    // Store tile into workspace, layout hp[b][head][node][HID]
    float* outp = hp + (((size_t)b * N_HEADS + hd) * NNODE) * HID;
#pragma unroll
    for (int nt = 0; nt < 4; ++nt) {
#pragma unroll
        for (int r = 0; r < 8; ++r) {
            const int node = rt * 16 + r + 8 * half;
            const int d    = nt * 16 + lm;
            outp[(size_t)node * HID + d] = acc[nt][r];
        }
    }
}

// ---------------------------------------------------------------------------
// Kernel 2: ei[r] = hp_row . a_i, ej[r] = hp_row . a_j. One wave per node row.
// ---------------------------------------------------------------------------
__global__ __launch_bounds__(256) void gat_edge_logits(
    const float* __restrict__ hp, const float* __restrict__ att_a,
    float* __restrict__ ei, float* __restrict__ ej)
{
    const int wave = threadIdx.x >> 5;
    const int lane = threadIdx.x & 31;
    const int row  = blockIdx.x * 8 + wave;      // (b*H + h)*NNODE + node
    const int head = (row / NNODE) % N_HEADS;

    const float* hr = hp + (size_t)row * HID;
    const float* aa = att_a + (size_t)head * (2 * HID);

    const float x0 = hr[lane], x1 = hr[lane + 32];
    float si = x0 * aa[lane]       + x1 * aa[lane + 32];
    float sj = x0 * aa[HID + lane] + x1 * aa[HID + lane + 32];
#pragma unroll
    for (int off = 16; off >= 1; off >>= 1) {
        si += __shfl_xor(si, off, 32);
        sj += __shfl_xor(sj, off, 32);
    }
    if (lane == 0) { ei[row] = si; ej[row] = sj; }
}

// ---------------------------------------------------------------------------
// Kernel 3: per logit row i: leaky(ei[i]+ej[j]) -> top-16 (+diag) -> softmax
// -> sparse gather of h_prime rows -> ELU. One wave per row, 16 j per lane.
// ---------------------------------------------------------------------------
__global__ __launch_bounds__(256) void gat_topk_softmax_agg(
    const float* __restrict__ hp, const float* __restrict__ ei,
    const float* __restrict__ ej, float* __restrict__ out)
{
    __shared__ int   s_cnt[8];
    __shared__ int   s_idx[8][64];
    __shared__ float s_p[8][64];

    const int wave = threadIdx.x >> 5;
    const int lane = threadIdx.x & 31;
    const int row  = blockIdx.x * 8 + wave;          // global row
    const int i    = row & (NNODE - 1);
    const int hb   = (row >> 9) << 9;                // head base row index
    const float* hpHead = hp + (size_t)hb * HID;

    const float eii = ei[row];
    float ev[16];
#pragma unroll
    for (int t = 0; t < 16; ++t) {
        const int j = t * 32 + lane;
        const float e = eii + ej[hb + j];
        ev[t] = e > 0.f ? e : LEAKY * e;
    }

    // --- top-16 via iterative max extraction (values assumed distinct) ---
    float prev = __builtin_inff();
    float maxv = -__builtin_inff();
    for (int it = 0; it < TOPK; ++it) {
        float lmax = -__builtin_inff();
#pragma unroll
        for (int t = 0; t < 16; ++t) {
            const float v = ev[t];
            if (v < prev && v > lmax) lmax = v;
        }
#pragma unroll
        for (int off = 16; off >= 1; off >>= 1) {
            const float o = __shfl_xor(lmax, off, 32);
            lmax = lmax > o ? lmax : o;
        }
        if (it == 0) maxv = lmax;
        prev = lmax;
    }
    const float thr = prev;                           // 16th largest

    if (lane == 0) s_cnt[wave] = 0;
    __syncthreads();

    // --- compact selected entries (top-k or diagonal) into LDS ---
#pragma unroll
    for (int t = 0; t < 16; ++t) {
        const int j = t * 32 + lane;
        if (ev[t] >= thr || j == i) {
            const int pos = atomicAdd(&s_cnt[wave], 1);
            if (pos < 64) {
                s_idx[wave][pos] = j;
                s_p[wave][pos]   = __expf(ev[t] - maxv);
            }
        }
    }
    __syncthreads();

    int m = s_cnt[wave]; if (m > 64) m = 64;
    float Z = 0.f;
    for (int q = 0; q < m; ++q) Z += s_p[wave][q];

    float a0 = 0.f, a1 = 0.f;                         // d = lane, lane+32
    for (int q = 0; q < m; ++q) {
        const int   j = s_idx[wave][q];
        const float p = s_p[wave][q];
        const float* r = hpHead + (size_t)j * HID;
        a0 += p * r[lane];
        a1 += p * r[lane + 32];
    }
    const float inv = 1.f / Z;
    a0 *= inv; a1 *= inv;
    a0 = a0 > 0.f ? a0 : __expf(a0) - 1.f;            // ELU
    a1 = a1 > 0.f ? a1 : __expf(a1) - 1.f;
    out[(size_t)row * HID + lane]      = a0;
    out[(size_t)row * HID + lane + 32] = a1;
}

// ---------------------------------------------------------------------------
extern "C" void kernel_launch(void* const* d_in, const int* in_sizes, int n_in,
                              void* d_out, int out_size, void* d_ws, size_t ws_size,
                              hipStream_t stream) {
    (void)in_sizes; (void)n_in; (void)out_size; (void)ws_size;
    const float* h     = (const float*)d_in[0];   // (32, 512, 128)
    const float* W     = (const float*)d_in[1];   // (128, 512)
    const float* att_a = (const float*)d_in[2];   // (8, 128)
    float* out = (float*)d_out;                   // (32, 8, 512, 64)

    float* hp = (float*)d_ws;                                        // 33.5 MB
    float* ei = hp + (size_t)BS * N_HEADS * NNODE * HID;             // 0.5 MB
    float* ej = ei + (size_t)BS * N_HEADS * NNODE;                   // 0.5 MB

    gat_gemm_hprime<<<dim3(BS, NNODE / 16), 256, 0, stream>>>(h, W, hp);
    gat_edge_logits<<<dim3(BS * N_HEADS * NNODE / 8), 256, 0, stream>>>(hp, att_a, ei, ej);
    gat_topk_softmax_agg<<<dim3(BS * N_HEADS * NNODE / 8), 256, 0, stream>>>(hp, ei, ej, out);
}